// GpuNufft_32074815766961
// MI455X (gfx1250) — compile-verified
//
#include <hip/hip_runtime.h>
#include <math.h>

// ---------------- problem constants (from setup_inputs) ----------------
#define HDIM 320
#define NC   16
#define NM   204800
#define OSF  2
#define GDIM 640               // OSF*HDIM
#define PAD  160               // (GDIM-HDIM)/2
#define KW   3
#define PI_F 3.14159265358979323846f

typedef float v2f __attribute__((ext_vector_type(2)));
typedef float v8f __attribute__((ext_vector_type(8)));

// D(16x16,f32) += A(16x4,f32) x B(4x16,f32) on the CDNA5 matrix pipe
__device__ __forceinline__ v8f wmma4(v2f a, v2f b, v8f c) {
  return __builtin_amdgcn_wmma_f32_16x16x4_f32(
      /*neg_a=*/false, a, /*neg_b=*/false, b,
      /*c_mod=*/(short)0, c, /*reuse_a=*/false, /*reuse_b=*/false);
}

// beta = pi*sqrt((kw/osf)^2*(osf-0.5)^2 - 0.8)  -- constant-folds at -O3
__device__ __forceinline__ float kb_beta() {
  const float a = (float)KW / (float)OSF;
  const float b = (float)OSF - 0.5f;
  return PI_F * sqrtf(a * a * b * b - 0.8f);
}

// deapodization 1-D factor: sin(z)/z == sinh(s)/s here (argument always < 0)
__device__ __forceinline__ float deapod1(int i) {
  float u  = ((float)i - (float)HDIM * 0.5f) / (float)GDIM;
  float w  = PI_F * (float)KW * u;
  float be = kb_beta();
  float s  = sqrtf(be * be - w * w);
  return sinhf(s) / s;
}

// Abramowitz & Stegun I0 (both branches; beta*sqrt(t) reaches ~6.5)
__device__ __forceinline__ float bessel_i0f(float x) {
  float ax = fabsf(x);
  if (ax < 3.75f) {
    float t = ax / 3.75f; t *= t;
    return 1.0f + t*(3.5156229f + t*(3.0899424f + t*(1.2067492f +
                 t*(0.2659732f + t*(0.0360768f + t*0.0045813f)))));
  } else {
    float t = 3.75f / ax;
    float p = 0.39894228f + t*(0.01328592f + t*(0.00225319f + t*(-0.00157565f +
              t*(0.00916281f + t*(-0.02057706f + t*(0.02635537f +
              t*(-0.01647633f + t*0.00392377f)))))));
    return p * __expf(ax) * rsqrtf(ax);
  }
}

__device__ __forceinline__ float kb_w(float d) {
  float r  = 2.0f * d / (float)KW;
  float tt = 1.0f - r * r;
  if (tt <= 0.0f) return 0.0f;
  return bessel_i0f(kb_beta() * sqrtf(tt)) * (1.0f / (float)KW);
}

// =====================================================================
// Kernel 1: deapodized coil images -> zero-padded, ifftshifted grid B
//   B[c][n1][n2] = grid[c][(n1+G/2)%G][(n2+G/2)%G]   (ifftshift folded)
// =====================================================================
__global__ void nufft_prep(const float* __restrict__ xin,
                           const float* __restrict__ csm,
                           float2* __restrict__ B) {
  int idx = blockIdx.x * blockDim.x + threadIdx.x;
  if (idx >= NC * GDIM * GDIM) return;
  int gx = idx % GDIM;
  int gy = (idx / GDIM) % GDIM;
  int c  = idx / (GDIM * GDIM);
  int sy = (gy + GDIM / 2) % GDIM;
  int sx = (gx + GDIM / 2) % GDIM;
  float2 v = make_float2(0.0f, 0.0f);
  if (sy >= PAD && sy < PAD + HDIM && sx >= PAD && sx < PAD + HDIM) {
    int y = sy - PAD, x = sx - PAD;
    float xr = xin[((size_t)y * HDIM + x) * 2 + 0];
    float xi = xin[((size_t)y * HDIM + x) * 2 + 1];
    float cr = csm[(((size_t)c * HDIM + y) * HDIM + x) * 2 + 0];
    float ci = csm[(((size_t)c * HDIM + y) * HDIM + x) * 2 + 1];
    float inva = 1.0f / (deapod1(y) * deapod1(x));
    v.x = (cr * xr - ci * xi) * inva;
    v.y = (cr * xi + ci * xr) * inva;
  }
  B[idx] = v;
}

// =====================================================================
// Kernel 2: 640-point FFT per line, four-step 640 = 16 * 40.
// One wave32 per line.  n = 40*n1 + n2,  k = k1 + 16*k2,  k2 = j1 + 5*j2.
//  stage 1 (WMMA): Y[k1][n2]  = sum_n1 W16^(n1*k1) * x[40*n1+n2]
//                  (complex 16x16x40 GEMM on v_wmma_f32_16x16x4_f32)
//  stage 2       : Z = Y * W640^(n2*k1)
//  stage 3       : 16 x DFT-40 (radix-5 + 3 Stockham radix-2 stages)
// Input line staged via async global->LDS DMA (ASYNCcnt).
// Row pass: stride_elem=1, stride_line=GDIM. Column pass: swapped.
// =====================================================================
__global__ __launch_bounds__(32)
void nufft_fft_line(float2* __restrict__ d, int stride_elem, int stride_line) {
  __shared__ float2 bufX[GDIM];            // input line (natural order)
  __shared__ float  zr[GDIM], zi[GDIM];    // stage-2 out / stage-3 pong
  __shared__ float  vr[GDIM], vi[GDIM];    // stage-3 ping
  const int t      = threadIdx.x;          // 0..31 (one wave)
  const int lane16 = t & 15;
  const int hi     = t >> 4;               // 0 | 1 (lane half)
  const int l      = blockIdx.x;           // coil*GDIM + line
  const int coil   = l / GDIM;
  const int line   = l % GDIM;
  const size_t base = (size_t)coil * GDIM * GDIM + (size_t)line * stride_line;

  // ---- async global -> LDS staging of the line -----------------------
  for (int i = 0; i < 20; ++i) {
    int j = t + 32 * i;
    const float2* gp = d + base + (size_t)j * stride_elem;
    unsigned lofs = (unsigned)(size_t)(&bufX[j]);    // low 32 bits = LDS offset
    asm volatile("global_load_async_to_lds_b64 %0, %1, off"
                 :: "v"(lofs), "v"(gp) : "memory");
  }

  // ---- W16 A-fragments: tile-invariant, built once -------------------
  // A layout (f32 16x4): element e ~ VGPR e; lane row M = lane16;
  // K = 4*kk + 2*hi + e.  W16 = e^{-i*2pi*M*K/16}: wr=cos, wip=Im=-sin.
  v2f awr[4], awip[4], awin[4];
  #pragma unroll
  for (int kk = 0; kk < 4; ++kk) {
    #pragma unroll
    for (int e = 0; e < 2; ++e) {
      int k = 4 * kk + 2 * hi + e;
      float sn, cs;
      __sincosf(-2.0f * PI_F * (float)(lane16 * k) / 16.0f, &sn, &cs);
      awr[kk][e]  = cs;                              //  Re(W16)
      awip[kk][e] = sn;                              //  Im(W16) = -sin
      awin[kk][e] = -sn;                             // -Im(W16)
    }
  }
  asm volatile("s_wait_asynccnt 0" ::: "memory");
  __syncthreads();

  // ---- stage 1+2: DFT-16 over n1 via WMMA, then W640 twiddle ---------
  // C/D layout: VGPR v ~ row M = v + 8*hi, column N = lane16
  for (int tile = 0; tile < 3; ++tile) {             // n2 tiles: 0,16,32
    const int n2 = tile * 16 + lane16;               // >=40 => zero pad
    v8f yr = {}; v8f yi = {};
    #pragma unroll
    for (int kk = 0; kk < 4; ++kk) {
      v2f bxr, bxi;
      #pragma unroll
      for (int e = 0; e < 2; ++e) {
        int k = 4 * kk + 2 * hi + e;                 // shared K index
        float2 xv = (n2 < 40) ? bufX[40 * k + n2] : make_float2(0.f, 0.f);
        bxr[e] = xv.x;  bxi[e] = xv.y;
      }
      yr = wmma4(awr[kk],  bxr, yr);                 // Yr += Wr*Xr
      yr = wmma4(awin[kk], bxi, yr);                 // Yr += (-Wi)*Xi
      yi = wmma4(awr[kk],  bxi, yi);                 // Yi += Wr*Xi
      yi = wmma4(awip[kk], bxr, yi);                 // Yi += Wi*Xr
    }
    if (n2 < 40) {
      #pragma unroll
      for (int v = 0; v < 8; ++v) {
        int k1 = v + 8 * hi;
        float sn, cs;                                // W640^{k1*n2}
        __sincosf(-2.0f * PI_F * (float)(k1 * n2) / (float)GDIM, &sn, &cs);
        float a = yr[v], b = yi[v];
        zr[k1 * 40 + n2] = a * cs - b * sn;
        zi[k1 * 40 + n2] = a * sn + b * cs;
      }
    }
  }
  __syncthreads();

  // ---- stage 3a: radix-5 over m1 (n2 = 8*m1 + m2) + W40 twiddle ------
  float c5[5], s5[5];
  #pragma unroll
  for (int j = 0; j < 5; ++j)
    __sincosf(-2.0f * PI_F * (float)j / 5.0f, &s5[j], &c5[j]);
  for (int task = t; task < 128; task += 32) {       // (k1, m2): 16*8 tasks
    int k1 = task >> 3, m2 = task & 7;
    float xr5[5], xi5[5];
    #pragma unroll
    for (int m1 = 0; m1 < 5; ++m1) {
      int idx = k1 * 40 + 8 * m1 + m2;
      xr5[m1] = zr[idx];  xi5[m1] = zi[idx];
    }
    #pragma unroll
    for (int j1 = 0; j1 < 5; ++j1) {
      float ar = 0.f, ai = 0.f;
      #pragma unroll
      for (int m1 = 0; m1 < 5; ++m1) {
        int j = (m1 * j1) % 5;
        ar += xr5[m1] * c5[j] - xi5[m1] * s5[j];
        ai += xr5[m1] * s5[j] + xi5[m1] * c5[j];
      }
      float sn, cs;                                  // W40^{m2*j1}
      __sincosf(-2.0f * PI_F * (float)(m2 * j1) / 40.0f, &sn, &cs);
      vr[k1 * 40 + j1 * 8 + m2] = ar * cs - ai * sn;
      vi[k1 * 40 + j1 * 8 + m2] = ar * sn + ai * cs;
    }
  }
  __syncthreads();

  // ---- stage 3b: 80 x 8-pt Stockham radix-2 (3 stages, autosort) -----
  float *sr = vr, *si = vi, *dr = zr, *di = zi;
  int half = 4, m = 1;
  for (int st = 0; st < 3; ++st) {
    for (int task = t; task < 320; task += 32) {     // 80 arrays * 4 bf
      int arr = task >> 2;                           // (k1,j1) pair
      int i   = task & 3;
      int b8  = (arr / 5) * 40 + (arr % 5) * 8;      // k1*40 + j1*8
      int p   = i / m, q = i - p * m;
      float axr = sr[b8 + q + m * p],          ayi = si[b8 + q + m * p];
      float bxr = sr[b8 + q + m * (p + half)], byi = si[b8 + q + m * (p + half)];
      float sn, cs;
      __sincosf(-PI_F * (float)p / (float)half, &sn, &cs);
      float dxr = axr - bxr, dyi = ayi - byi;
      dr[b8 + q + m * (2 * p)]     = axr + bxr;
      di[b8 + q + m * (2 * p)]     = ayi + byi;
      dr[b8 + q + m * (2 * p + 1)] = dxr * cs - dyi * sn;
      di[b8 + q + m * (2 * p + 1)] = dxr * sn + dyi * cs;
    }
    __syncthreads();
    float* tp;
    tp = sr; sr = dr; dr = tp;
    tp = si; si = di; di = tp;
    half >>= 1; m <<= 1;
  }
  // final data in sr/si: storage s = k1*40 + j1*8 + j2  ->  k = k1+16*j1+80*j2

  // ---- store ---------------------------------------------------------
  for (int s = t; s < GDIM; s += 32) {
    int k1 = s / 40;
    int r  = s - k1 * 40;
    int j1 = r >> 3, j2 = r & 7;
    int n  = k1 + 16 * j1 + 80 * j2;
    d[base + (size_t)n * stride_elem] = make_float2(sr[s], si[s]);
  }
}

// =====================================================================
// Kernel 3: 3x3 Kaiser-Bessel degridding + sqrt(dcf).
// fftshift of the spectrum and the 1/G FFT normalization are folded in.
// =====================================================================
__global__ void nufft_degrid(const float2* __restrict__ ksp,
                             const float* __restrict__ traj,
                             const float* __restrict__ dcf,
                             float* __restrict__ out) {
  int mm = blockIdx.x * blockDim.x + threadIdx.x;
  if (mm >= NM) return;
  float py = traj[mm]      * (float)GDIM + (float)GDIM * 0.5f;
  float px = traj[NM + mm] * (float)GDIM + (float)GDIM * 0.5f;
  float cy = rintf(py), cx = rintf(px);              // ties-to-even == jnp.rint
  float wy[3], wx[3];
  int   iy[3], ix[3];
  #pragma unroll
  for (int o = 0; o < 3; ++o) {
    float ny = cy + (float)(o - 1);
    wy[o] = kb_w(ny - py);
    int ty = (int)ny % GDIM; if (ty < 0) ty += GDIM;
    iy[o] = (ty + GDIM / 2) % GDIM;                  // fftshift folded
    float nx = cx + (float)(o - 1);
    wx[o] = kb_w(nx - px);
    int tx = (int)nx % GDIM; if (tx < 0) tx += GDIM;
    ix[o] = (tx + GDIM / 2) % GDIM;
  }
  float sd = sqrtf(dcf[mm]) * (1.0f / (float)GDIM);  // sqrt(dcf) * 1/G
  #pragma unroll 1
  for (int c = 0; c < NC; ++c) {
    const float2* basep = ksp + (size_t)c * GDIM * GDIM;
    float ar = 0.0f, ai = 0.0f;
    #pragma unroll
    for (int a = 0; a < 3; ++a) {
      const float2* rowp = basep + (size_t)iy[a] * GDIM;
      float wya = wy[a];
      #pragma unroll
      for (int b = 0; b < 3; ++b) {
        float w  = wya * wx[b];
        float2 v = rowp[ix[b]];
        ar += v.x * w;
        ai += v.y * w;
      }
    }
    out[((size_t)c * NM + mm) * 2 + 0] = ar * sd;
    out[((size_t)c * NM + mm) * 2 + 1] = ai * sd;
  }
}

// =====================================================================
extern "C" void kernel_launch(void* const* d_in, const int* in_sizes, int n_in,
                              void* d_out, int out_size, void* d_ws, size_t ws_size,
                              hipStream_t stream) {
  (void)in_sizes; (void)n_in; (void)out_size; (void)ws_size;
  const float* x    = (const float*)d_in[0];   // [320,320,2]
  const float* csm  = (const float*)d_in[1];   // [16,320,320,2]
  const float* traj = (const float*)d_in[2];   // [2,204800]
  const float* dcf  = (const float*)d_in[3];   // [204800]
  // d_in[4..6] = img_dim/osf/kernel_width scalars, baked as constants
  float2* B   = (float2*)d_ws;                 // [16,640,640] complex, 52.4 MB
  float*  out = (float*)d_out;                 // [16,204800,2]

  {
    int n = NC * GDIM * GDIM;
    nufft_prep<<<(n + 255) / 256, 256, 0, stream>>>(x, csm, B);
  }
  // row FFT then column FFT (in place; one wave32 owns one line)
  nufft_fft_line<<<NC * GDIM, 32, 0, stream>>>(B, 1, GDIM);
  nufft_fft_line<<<NC * GDIM, 32, 0, stream>>>(B, GDIM, 1);
  nufft_degrid<<<(NM + 255) / 256, 256, 0, stream>>>(B, traj, dcf, out);
}